// ensembleThreeModel_5128190951894
// MI455X (gfx1250) — compile-verified
//
#include <hip/hip_runtime.h>
#include <stdint.h>

// Problem constants (from reference): B=65536 rows, C=1000 classes, fp32.
#define BATCH    65536
#define NCLS     1000
#define ROWS     4        // rows (waves) per workgroup
#define NTHREADS 128      // 4 wave32 waves

typedef unsigned int        u32;
typedef unsigned long long  u64;
typedef __attribute__((ext_vector_type(4))) u32   v4u;
typedef __attribute__((ext_vector_type(8))) int   v8i;
typedef __attribute__((ext_vector_type(4))) int   v4i;
typedef __attribute__((ext_vector_type(4))) float f4;

#if defined(__AMDGCN__) && __has_builtin(__builtin_amdgcn_tensor_load_to_lds)
#define USE_TDM 1
#else
#define USE_TDM 0
#endif

#if USE_TDM
// Build a 2-D Tensor-DMA descriptor (D#) for a ROWS x NCLS f32 tile and issue
// TENSOR_LOAD_TO_LDS.  Bitfields per CDNA5 ISA
// sections 8.3 (group0) and 8.4 (group1).  Groups 2/3 zero (2-D tensor).
// This toolchain exposes the 6-arg form:
//   (uint32x4 g0, int32x8 g1, int32x4 g2, int32x4 g3, int32x8 extra, i32 cpol)
__device__ __forceinline__ void tdm_load_tile(const float* gptr, u32 lds_byte_off)
{
  u64 ga = (u64)(uintptr_t)gptr;
  v4u g0;
  g0.x = 0x1u;                                    // count=1, user descriptor
  g0.y = lds_byte_off;                            // lds_addr (bytes)
  g0.z = (u32)(ga & 0xFFFFFFFFull);               // global_addr[31:0]
  g0.w = (u32)((ga >> 32) & 0x1FFFFFFull)         // global_addr[56:32]
       | 0x80000000u;                             // type=2 ("image")
  v8i g1;
  g1[0] = (int)(2u << 16);                        // wg_mask=0, data_size=2 (4B)
  g1[1] = (int)(((u32)NCLS & 0xFFFFu) << 16);     // tensor_dim0[15:0]
  g1[2] = (int)((((u32)NCLS >> 16) & 0xFFFFu)     // tensor_dim0[31:16]
       | (((u32)ROWS & 0xFFFFu) << 16));          // tensor_dim1[15:0]
  g1[3] = (int)(((u32)NCLS & 0xFFFFu) << 16);     // tensor_dim1[31:16]=0 | tile_dim0=NCLS
  g1[4] = (int)((u32)ROWS & 0xFFFFu);             // tile_dim1=ROWS, tile_dim2=0
  g1[5] = (int)NCLS;                              // tensor_dim0_stride[31:0]
  g1[6] = 0;                                      // stride0[47:32]=0, dim1_stride lo=0
  g1[7] = 0;
  v4i z4 = {0, 0, 0, 0};
  v8i z8 = {0, 0, 0, 0, 0, 0, 0, 0};
  __builtin_amdgcn_tensor_load_to_lds(g0, g1, z4, z4, z8, /*cpol=*/0);
}
#endif

// Per-lane argmax update over 4 consecutive columns (strict > keeps the
// first-occurrence / lowest-index semantics of jnp.argmax within a lane,
// since each lane visits its columns in ascending order).
__device__ __forceinline__ void amax4(f4 x, int col, float& v, int& idx)
{
  if (x.x > v) { v = x.x; idx = col + 0; }
  if (x.y > v) { v = x.y; idx = col + 1; }
  if (x.z > v) { v = x.z; idx = col + 2; }
  if (x.w > v) { v = x.w; idx = col + 3; }
}

// Cross-lane (wave32) argmax butterfly: max value, lowest index on ties.
__device__ __forceinline__ void wreduce(float& v, int& idx)
{
#pragma unroll
  for (int off = 16; off > 0; off >>= 1) {
    float ov = __shfl_xor(v,   off, 32);
    int   oi = __shfl_xor(idx, off, 32);
    if (ov > v || (ov == v && oi < idx)) { v = ov; idx = oi; }
  }
}

__global__ __launch_bounds__(NTHREADS)
void ensemble_vote_kernel(const float* __restrict__ o1,
                          const float* __restrict__ o2,
                          const float* __restrict__ o3,
                          float* __restrict__ out)
{
  __shared__ float smem[3 * ROWS * NCLS];   // 48 KB: tile of all three inputs
  const int tid  = threadIdx.x;
  const int wave = tid >> 5;
  const int lane = tid & 31;
  const int row0 = blockIdx.x * ROWS;

#if USE_TDM
  if (tid < 32) {                           // wave 0 drives the TDM
    u32 lds0 = (u32)(uintptr_t)(&smem[0]);  // low 32 bits = LDS byte offset
    tdm_load_tile(o1 + (size_t)row0 * NCLS, lds0);
    tdm_load_tile(o2 + (size_t)row0 * NCLS, lds0 + (u32)(ROWS * NCLS * 4));
    tdm_load_tile(o3 + (size_t)row0 * NCLS, lds0 + (u32)(2 * ROWS * NCLS * 4));
    __builtin_amdgcn_s_wait_tensorcnt(0);
  }
#else
  { // cooperative float4 global->LDS copy fallback (host pass / no-TDM toolchain)
    const f4* g1 = (const f4*)(o1 + (size_t)row0 * NCLS);
    const f4* g2 = (const f4*)(o2 + (size_t)row0 * NCLS);
    const f4* g3 = (const f4*)(o3 + (size_t)row0 * NCLS);
    f4* l1 = (f4*)&smem[0];
    f4* l2 = (f4*)&smem[ROWS * NCLS];
    f4* l3 = (f4*)&smem[2 * ROWS * NCLS];
    for (int i = tid; i < ROWS * NCLS / 4; i += NTHREADS) {
      l1[i] = g1[i]; l2[i] = g2[i]; l3[i] = g3[i];
    }
  }
#endif
  __syncthreads();

  const int    row = row0 + wave;
  const float* s1  = &smem[wave * NCLS];
  const float* s2  = &smem[ROWS * NCLS + wave * NCLS];
  const float* s3  = &smem[2 * ROWS * NCLS + wave * NCLS];

  // ---- Pass 1: per-row argmax of each model (one wave per row) ----
  float v1 = -1.0f, v2 = -1.0f, v3 = -1.0f;   // probabilities are > 0
  int   i1 = 0, i2 = 0, i3 = 0;
#pragma unroll
  for (int k = 0; k < 8; ++k) {
    int col = k * 128 + lane * 4;             // cols 0..996 step 4, disjoint per lane
    if (col <= NCLS - 4) {
      f4 a = *(const f4*)(s1 + col);
      f4 b = *(const f4*)(s2 + col);
      f4 c = *(const f4*)(s3 + col);
      amax4(a, col, v1, i1);
      amax4(b, col, v2, i2);
      amax4(c, col, v3, i3);
    }
  }
  wreduce(v1, i1);
  wreduce(v2, i2);
  wreduce(v3, i3);

  // ---- Majority vote (identical on all lanes) ----
  const bool eq12 = (i1 == i2), eq13 = (i1 == i3), eq23 = (i2 == i3);
  const int  cmax = (eq12 && eq13) ? 3 : ((eq12 || eq13 || eq23) ? 2 : 1);
  const int  maj  = (eq12 || eq13) ? i1 : (eq23 ? i2 : i1);
  const int  value = (cmax < 2) ? i3 : maj;

  const float m1 = (i1 == value) ? 1.0f : 0.0f;
  const float m2 = (i2 == value) ? 1.0f : 0.0f;
  const float m3 = (i3 == value) ? 1.0f : 0.0f;
  const float invd = 1.0f / (float)cmax;
  const float third = 1.0f / 3.0f;

  // Output layout (flat, return order): pred [B] | avg_correct [B*C] | avg [B*C]
  if (lane == 0) out[row] = (float)value;     // class index (exact in f32)

  float* outc = out + BATCH + (size_t)row * NCLS;
  float* outa = out + BATCH + (size_t)BATCH * NCLS + (size_t)row * NCLS;

  // ---- Pass 2: fused elementwise outputs from the LDS tile ----
#pragma unroll
  for (int k = 0; k < 8; ++k) {
    int col = k * 128 + lane * 4;
    if (col <= NCLS - 4) {
      f4 a = *(const f4*)(s1 + col);
      f4 b = *(const f4*)(s2 + col);
      f4 c = *(const f4*)(s3 + col);
      f4 avgc, avg;
      avgc.x = (a.x * m1 + b.x * m2 + c.x * m3) * invd;
      avgc.y = (a.y * m1 + b.y * m2 + c.y * m3) * invd;
      avgc.z = (a.z * m1 + b.z * m2 + c.z * m3) * invd;
      avgc.w = (a.w * m1 + b.w * m2 + c.w * m3) * invd;
      avg.x  = (a.x + b.x + c.x) * third;
      avg.y  = (a.y + b.y + c.y) * third;
      avg.z  = (a.z + b.z + c.z) * third;
      avg.w  = (a.w + b.w + c.w) * third;
      // streaming outputs: never re-read -> non-temporal b128 stores
      __builtin_nontemporal_store(avgc, (f4*)(outc + col));
      __builtin_nontemporal_store(avg,  (f4*)(outa + col));
    }
  }
}

extern "C" void kernel_launch(void* const* d_in, const int* in_sizes, int n_in,
                              void* d_out, int out_size, void* d_ws, size_t ws_size,
                              hipStream_t stream)
{
  (void)in_sizes; (void)n_in; (void)out_size; (void)d_ws; (void)ws_size;
  const float* o1 = (const float*)d_in[0];
  const float* o2 = (const float*)d_in[1];
  const float* o3 = (const float*)d_in[2];
  float* out = (float*)d_out;

  dim3 grid(BATCH / ROWS);   // 16384 blocks, 4 rows each
  dim3 block(NTHREADS);      // 4 wave32 waves, one row per wave
  ensemble_vote_kernel<<<grid, block, 0, stream>>>(o1, o2, o3, out);
}